// AdaptedEnzymeModel_14061722927712
// MI455X (gfx1250) — compile-verified
//
#include <hip/hip_runtime.h>

// ---------------------------------------------------------------------------
// Fused GNN-style model for MI455X (gfx1250, wave32).
// All GEMMs use V_WMMA_F32_16X16X4_F32 (exact fp32, RNE) so results match the
// fp32 JAX reference closely. One wave owns a 16-row tile and runs the whole
// layer chain out of LDS ping-pong buffers.
// ---------------------------------------------------------------------------

typedef __attribute__((ext_vector_type(2))) float v2f;
typedef __attribute__((ext_vector_type(8))) float v8f;

#define ASTRIDE 132   // 16-row activation tile row stride (floats); 132%64=4 -> conflict-free
#define ROWS    16    // rows (M) per wave tile
#define WPB     2     // waves per block
#define EPS     1e-5f

__device__ __forceinline__ float bn_apply(float h, const float* __restrict__ bn,
                                          int c, int C) {
  // bn is stacked [gamma, beta, mean, var] x C
  const float g  = bn[c];
  const float b  = bn[C + c];
  const float mu = bn[2 * C + c];
  const float vr = bn[3 * C + c];
  return (h - mu) * rsqrtf(vr + EPS) * g + b;
}

// Wave-level GEMM: out[16xN] = act_in[16xK] @ W[KxN] (+bias, optional relu, optional BN)
// act_in / act_out are wave-private LDS tiles with row stride ASTRIDE.
template <int K, int N, bool RELU, bool BN>
__device__ __forceinline__ void wave_gemm(const float* act_in, float* act_out,
                                          const float* __restrict__ W,
                                          const float* __restrict__ bias,
                                          const float* __restrict__ bn,
                                          int lane) {
  __builtin_prefetch(W, 0, 1);                 // global_prefetch_b8 (L2-resident weights)
  const int mA   = lane & 15;                  // A-frag: M = lane&15
  const int kOff = (lane >> 4) * 2;            // lanes 16-31 hold K+2,K+3
  const int n    = lane & 15;                  // B/C-frag: N = lane&15

#pragma unroll
  for (int nt = 0; nt < N / 16; ++nt) {
    v8f acc = {0.f, 0.f, 0.f, 0.f, 0.f, 0.f, 0.f, 0.f};
#pragma unroll
    for (int k0 = 0; k0 < K; k0 += 4) {
      v2f a, b;
      a.x = act_in[mA * ASTRIDE + k0 + kOff];
      a.y = act_in[mA * ASTRIDE + k0 + kOff + 1];
      b.x = W[(k0 + kOff) * N + nt * 16 + n];
      b.y = W[(k0 + kOff + 1) * N + nt * 16 + n];
      acc = __builtin_amdgcn_wmma_f32_16x16x4_f32(false, a, false, b,
                                                  (short)0, acc, false, false);
    }
    const int   nc = nt * 16 + n;
    const float bv = bias[nc];
    float g = 0.f, bb = 0.f, mu = 0.f, iv = 0.f;
    if (BN) {
      g  = bn[nc];
      bb = bn[N + nc];
      mu = bn[2 * N + nc];
      iv = rsqrtf(bn[3 * N + nc] + EPS);
    }
#pragma unroll
    for (int j = 0; j < 8; ++j) {              // C/D: M = j + 8*(lane>>4), N = lane&15
      const int m = j + 8 * (lane >> 4);
      float v = acc[j] + bv;
      if (RELU) v = fmaxf(v, 0.f);
      if (BN)   v = (v - mu) * iv * g + bb;
      act_out[m * ASTRIDE + nc] = v;
    }
  }
}

// ---------------------------------------------------------------------------
// Kernel 1: per-element MLP chain fused with segment-sum accumulation.
// ---------------------------------------------------------------------------
__global__ void __launch_bounds__(WPB * 32)
node_kernel(const float* __restrict__ x, const int* __restrict__ batch, int T,
            const float* __restrict__ ne_w1, const float* __restrict__ ne_b1,
            const float* __restrict__ ne_bn1,
            const float* __restrict__ ne_w2, const float* __restrict__ ne_b2,
            const float* __restrict__ ne_bn2,
            const float* __restrict__ c1a_w, const float* __restrict__ c1a_b,
            const float* __restrict__ c1b_w, const float* __restrict__ c1b_b,
            const float* __restrict__ cbn1,
            const float* __restrict__ c2a_w, const float* __restrict__ c2a_b,
            const float* __restrict__ c2b_w, const float* __restrict__ c2b_b,
            const float* __restrict__ cbn2,
            float* __restrict__ sums, float* __restrict__ cnts) {
  __shared__ float s_act[WPB][2][ROWS * ASTRIDE];
  const int  lane     = threadIdx.x & 31;
  const int  wave     = threadIdx.x >> 5;
  const long tileBase = (long)blockIdx.x * (WPB * ROWS) + (long)wave * ROWS;
  float* bufA = &s_act[wave][0][0];
  float* bufB = &s_act[wave][1][0];

  // Layer 1 (1 -> 32): outer product, relu + BN, done on VALU.
  for (int i = lane; i < ROWS * 32; i += 32) {
    const int  m   = i >> 5;
    const int  c   = i & 31;
    const long row = tileBase + m;
    const float xv = (row < (long)T) ? x[row] : 0.f;
    float h = fmaxf(xv * ne_w1[c] + ne_b1[c], 0.f);
    bufA[m * ASTRIDE + c] = bn_apply(h, ne_bn1, c, 32);
  }

  wave_gemm<32,  64,  true, true >(bufA, bufB, ne_w2, ne_b2, ne_bn2, lane);
  wave_gemm<64,  64,  true, false>(bufB, bufA, c1a_w, c1a_b, nullptr, lane);
  wave_gemm<64,  64,  true, true >(bufA, bufB, c1b_w, c1b_b, cbn1,    lane);
  wave_gemm<64,  128, true, false>(bufB, bufA, c2a_w, c2a_b, nullptr, lane);
  wave_gemm<128, 128, true, true >(bufA, bufB, c2b_w, c2b_b, cbn2,    lane);

  // Segment accumulation (batch is sorted -> localized contention).
  for (int m = 0; m < ROWS; ++m) {
    const long row = tileBase + m;
    if (row >= (long)T) break;
    const int seg = batch[row];
    for (int c = lane; c < 128; c += 32) {
      __hip_atomic_fetch_add(&sums[(long)seg * 128 + c], bufB[m * ASTRIDE + c],
                             __ATOMIC_RELAXED, __HIP_MEMORY_SCOPE_AGENT);
    }
    if (lane == 0) {
      __hip_atomic_fetch_add(&cnts[seg], 1.f,
                             __ATOMIC_RELAXED, __HIP_MEMORY_SCOPE_AGENT);
    }
  }
}

// ---------------------------------------------------------------------------
// Kernel 2: per-graph head (mean -> 128 -> 64 -> 32 -> 7).
// ---------------------------------------------------------------------------
__global__ void __launch_bounds__(WPB * 32)
head_kernel(const float* __restrict__ sums, const float* __restrict__ cnts, int B,
            const float* __restrict__ f1_w, const float* __restrict__ f1_b,
            const float* __restrict__ fbn1,
            const float* __restrict__ f2_w, const float* __restrict__ f2_b,
            const float* __restrict__ fbn2,
            const float* __restrict__ f3_w, const float* __restrict__ f3_b,
            float* __restrict__ out) {
  __shared__ float s_act[WPB][2][ROWS * ASTRIDE];
  const int  lane = threadIdx.x & 31;
  const int  wave = threadIdx.x >> 5;
  const long base = (long)blockIdx.x * (WPB * ROWS) + (long)wave * ROWS;
  float* bufA = &s_act[wave][0][0];
  float* bufB = &s_act[wave][1][0];

  // g = sums / max(cnts, 1)
  for (int i = lane; i < ROWS * 128; i += 32) {
    const int  m   = i >> 7;
    const int  c   = i & 127;
    const long gph = base + m;
    float v = 0.f;
    if (gph < (long)B) v = sums[gph * 128 + c] / fmaxf(cnts[gph], 1.f);
    bufA[m * ASTRIDE + c] = v;
  }

  wave_gemm<128, 64, true, true>(bufA, bufB, f1_w, f1_b, fbn1, lane);
  wave_gemm<64,  32, true, true>(bufB, bufA, f2_w, f2_b, fbn2, lane);

  // Final 32 -> 7 (N padded to a 16-wide WMMA tile; cols >= 7 fed zeros).
  const int kOff = (lane >> 4) * 2;
  const int n    = lane & 15;
  v8f acc = {0.f, 0.f, 0.f, 0.f, 0.f, 0.f, 0.f, 0.f};
#pragma unroll
  for (int k0 = 0; k0 < 32; k0 += 4) {
    v2f a, b;
    a.x = bufA[(lane & 15) * ASTRIDE + k0 + kOff];
    a.y = bufA[(lane & 15) * ASTRIDE + k0 + kOff + 1];
    b.x = (n < 7) ? f3_w[(k0 + kOff) * 7 + n] : 0.f;
    b.y = (n < 7) ? f3_w[(k0 + kOff + 1) * 7 + n] : 0.f;
    acc = __builtin_amdgcn_wmma_f32_16x16x4_f32(false, a, false, b,
                                                (short)0, acc, false, false);
  }
  if (n < 7) {
    const float bv = f3_b[n];
#pragma unroll
    for (int j = 0; j < 8; ++j) {
      const int  m   = j + 8 * (lane >> 4);
      const long gph = base + m;
      if (gph < (long)B) out[gph * 7 + n] = acc[j] + bv;
    }
  }
}

__global__ void zero_kernel(float* __restrict__ p, long n) {
  const long i = (long)blockIdx.x * blockDim.x + threadIdx.x;
  if (i < n) p[i] = 0.f;
}

// ---------------------------------------------------------------------------
extern "C" void kernel_launch(void* const* d_in, const int* in_sizes, int n_in,
                              void* d_out, int out_size, void* d_ws, size_t ws_size,
                              hipStream_t stream) {
  const float* x      = (const float*)d_in[0];
  const int*   batch  = (const int*)d_in[1];
  // d_in[2] = num_graphs (device scalar); recover B on host from out_size.
  const int T = in_sizes[0];
  const int B = out_size / 7;

  const float* ne_w1  = (const float*)d_in[3];
  const float* ne_b1  = (const float*)d_in[4];
  const float* ne_bn1 = (const float*)d_in[5];
  const float* ne_w2  = (const float*)d_in[6];
  const float* ne_b2  = (const float*)d_in[7];
  const float* ne_bn2 = (const float*)d_in[8];
  const float* c1a_w  = (const float*)d_in[9];
  const float* c1a_b  = (const float*)d_in[10];
  const float* c1b_w  = (const float*)d_in[11];
  const float* c1b_b  = (const float*)d_in[12];
  const float* cbn1   = (const float*)d_in[13];
  const float* c2a_w  = (const float*)d_in[14];
  const float* c2a_b  = (const float*)d_in[15];
  const float* c2b_w  = (const float*)d_in[16];
  const float* c2b_b  = (const float*)d_in[17];
  const float* cbn2   = (const float*)d_in[18];
  const float* f1_w   = (const float*)d_in[19];
  const float* f1_b   = (const float*)d_in[20];
  const float* fbn1   = (const float*)d_in[21];
  const float* f2_w   = (const float*)d_in[22];
  const float* f2_b   = (const float*)d_in[23];
  const float* fbn2   = (const float*)d_in[24];
  const float* f3_w   = (const float*)d_in[25];
  const float* f3_b   = (const float*)d_in[26];

  float* sums = (float*)d_ws;                    // B x 128
  float* cnts = sums + (size_t)B * 128;          // B

  const long nz = (long)B * 128 + B;
  zero_kernel<<<(int)((nz + 255) / 256), 256, 0, stream>>>(sums, nz);

  const int rowsPerBlock = WPB * ROWS;
  node_kernel<<<(T + rowsPerBlock - 1) / rowsPerBlock, WPB * 32, 0, stream>>>(
      x, batch, T,
      ne_w1, ne_b1, ne_bn1, ne_w2, ne_b2, ne_bn2,
      c1a_w, c1a_b, c1b_w, c1b_b, cbn1,
      c2a_w, c2a_b, c2b_w, c2b_b, cbn2,
      sums, cnts);

  head_kernel<<<(B + rowsPerBlock - 1) / rowsPerBlock, WPB * 32, 0, stream>>>(
      sums, cnts, B,
      f1_w, f1_b, fbn1, f2_w, f2_b, fbn2, f3_w, f3_b,
      (float*)d_out);
}